// GCNBlock_35416300323759
// MI455X (gfx1250) — compile-verified
//
#include <hip/hip_runtime.h>
#include <hip/hip_bf16.h>
#include <math.h>

typedef __attribute__((ext_vector_type(2))) float v2f;
typedef __attribute__((ext_vector_type(8))) float v8f;

#define DCH 128          // feature dim
#define NGRAPH 64
#define GN_EPS 1e-5f
#define TILE_M 16        // rows per GEMM block
#define SX_LD (DCH + 4)  // LDS pad: stride 132 -> banks spread, 16B aligned

// ---------------------------------------------------------------------------
// K0: zero fill
// ---------------------------------------------------------------------------
__global__ void k_zero_f32(float* __restrict__ p, size_t n) {
    size_t i = (size_t)blockIdx.x * blockDim.x + threadIdx.x;
    if (i < n) p[i] = 0.0f;
}

// ---------------------------------------------------------------------------
// K-pre: Wt[n][k] = W[k][n]  (one-time 64KB transpose so GEMM B-fragments
// become contiguous b64 loads along K)
// grid: 64 blocks of 256 (8x8 tiles of 16x16)
// ---------------------------------------------------------------------------
__global__ __launch_bounds__(256) void k_transpose_w(
    const float* __restrict__ W, float* __restrict__ Wt) {
    __shared__ float tile[16][17];
    int bx = blockIdx.x & 7;
    int by = blockIdx.x >> 3;
    int tx = threadIdx.x & 15;
    int ty = threadIdx.x >> 4;
    tile[ty][tx] = W[(size_t)(by * 16 + ty) * DCH + bx * 16 + tx]; // coalesced read
    __syncthreads();
    Wt[(size_t)(bx * 16 + ty) * DCH + by * 16 + tx] = tile[tx][ty]; // coalesced write
}

// ---------------------------------------------------------------------------
// K1: h = x @ W via V_WMMA_F32_16X16X4_F32 (exact fp32 path)
// Block: 256 threads = 8 waves, one 16-row M-tile; wave w owns cols [16w,16w+16).
// A staged in padded LDS (ds_load_2addr_b64); B from transposed W (b64 loads,
// 64KB -> L0/L2 resident).
// ---------------------------------------------------------------------------
__global__ __launch_bounds__(256) void k_gemm_wmma(
    const float* __restrict__ x, const float* __restrict__ Wt,
    float* __restrict__ h, int N) {
    __shared__ float sX[TILE_M][SX_LD];

    const int t = threadIdx.x;
    const int rowBase = blockIdx.x * TILE_M;

    for (int i = t; i < TILE_M * DCH; i += 256) {
        int r = i >> 7;
        int c = i & (DCH - 1);
        int gr = rowBase + r;
        sX[r][c] = (gr < N) ? x[(size_t)gr * DCH + c] : 0.0f;
    }
    __syncthreads();

    const int lane = t & 31;
    const int wave = t >> 5;
    const int m    = lane & 15;          // A-row this lane holds
    const int koff = (lane >> 4) << 1;   // 0 for lanes 0-15, 2 for lanes 16-31
    const int n    = (wave << 4) + (lane & 15);  // output column

    const float* wrow = Wt + (size_t)n * DCH;    // row n of W^T = column n of W

    v8f acc = {};
#pragma unroll
    for (int kt = 0; kt < DCH; kt += 4) {
        // A fragment: 16x4 fp32 (ISA 7.12.2 layout), 8B LDS load
        v2f a = *reinterpret_cast<const v2f*>(&sX[m][kt + koff]);
        // B fragment: 4x16 fp32; W[k][n],W[k+1][n] contiguous in Wt -> b64 load
        v2f b = *reinterpret_cast<const v2f*>(wrow + kt + koff);
        acc = __builtin_amdgcn_wmma_f32_16x16x4_f32(
            false, a, false, b, (short)0, acc, false, false);
    }

    // D layout: VGPR v -> M = v (lanes 0-15) or v+8 (lanes 16-31), N = lane&15
    const int rhalf = (lane >> 4) << 3;
#pragma unroll
    for (int v = 0; v < 8; ++v) {
        int gr = rowBase + rhalf + v;
        if (gr < N) h[(size_t)gr * DCH + n] = acc[v];
    }
}

// ---------------------------------------------------------------------------
// K2: agg[dst] += h[src] * attr.  One wave per 4 edges; lane owns 4 channels.
// Edge metadata scalarized (readfirstlane -> SMEM); h/agg are L2-resident
// (51.2MB each << 192MB L2); hardware fp32 atomics.
// ---------------------------------------------------------------------------
#define EDGES_PER_WAVE 4
__global__ __launch_bounds__(256) void k_edge_scatter(
    const int* __restrict__ src, const int* __restrict__ dst,
    const float* __restrict__ attr, const float* __restrict__ h,
    float* __restrict__ agg, int E) {
    const int lane  = threadIdx.x & 31;
    const int ebase = (blockIdx.x * 8 + (threadIdx.x >> 5)) * EDGES_PER_WAVE;

    // speculative stream prefetch (safe: dropped on translation failure)
    __builtin_prefetch(&src[ebase + 4096], 0, 0);
    __builtin_prefetch(&dst[ebase + 4096], 0, 0);
    __builtin_prefetch(&attr[ebase + 4096], 0, 0);

#pragma unroll
    for (int j = 0; j < EDGES_PER_WAVE; ++j) {
        int e = ebase + j;
        if (e >= E) break;
        e = __builtin_amdgcn_readfirstlane(e);   // wave-uniform -> scalar loads
        int   s = src[e];
        int   d = dst[e];
        float a = attr[e];
        const float4* hv = reinterpret_cast<const float4*>(h + (size_t)s * DCH);
        float4 v = hv[lane];
        float* outp = agg + (size_t)d * DCH + lane * 4;
        unsafeAtomicAdd(outp + 0, v.x * a);
        unsafeAtomicAdd(outp + 1, v.y * a);
        unsafeAtomicAdd(outp + 2, v.z * a);
        unsafeAtomicAdd(outp + 3, v.w * a);
    }
}

// ---------------------------------------------------------------------------
// K3: h = relu(agg + b) in place; per-graph channel sums + node counts.
// batch_index is sorted -> run-length accumulate, flush on graph change.
// Block: 128 threads (one per channel) x 64 rows.
// ---------------------------------------------------------------------------
#define GROWS 64
__global__ __launch_bounds__(128) void k_bias_relu_mean(
    float* __restrict__ out, const float* __restrict__ bias,
    const int* __restrict__ batch, float* __restrict__ mean_sum,
    float* __restrict__ cnt, int N) {
    __shared__ int sg[GROWS];
    const int base = blockIdx.x * GROWS;
    const int c = threadIdx.x;
    int rows = N - base;
    if (rows > GROWS) rows = GROWS;
    for (int i = c; i < rows; i += 128) sg[i] = batch[base + i];
    __syncthreads();

    const float bv = bias[c];
    float acc = 0.0f, cacc = 0.0f;
    int curg = -1;
    for (int r = 0; r < rows; ++r) {
        int g = sg[r];
        if (g != curg) {
            if (curg >= 0) {
                unsafeAtomicAdd(&mean_sum[(size_t)curg * DCH + c], acc);
                if (c == 0) unsafeAtomicAdd(&cnt[curg], cacc);
            }
            acc = 0.0f; cacc = 0.0f; curg = g;
        }
        size_t idx = (size_t)(base + r) * DCH + c;
        float v = out[idx] + bv;
        v = v > 0.0f ? v : 0.0f;
        out[idx] = v;
        acc += v;
        cacc += 1.0f;
    }
    if (curg >= 0) {
        unsafeAtomicAdd(&mean_sum[(size_t)curg * DCH + c], acc);
        if (c == 0) unsafeAtomicAdd(&cnt[curg], cacc);
    }
}

// ---------------------------------------------------------------------------
// K4: per-graph variance sums of (h - mean*mean_scale), same run-length trick
// ---------------------------------------------------------------------------
__global__ __launch_bounds__(128) void k_var_pass(
    const float* __restrict__ out, const int* __restrict__ batch,
    const float* __restrict__ mean_sum, const float* __restrict__ cnt,
    const float* __restrict__ ms, float* __restrict__ var_sum, int N) {
    __shared__ int sg[GROWS];
    const int base = blockIdx.x * GROWS;
    const int c = threadIdx.x;
    int rows = N - base;
    if (rows > GROWS) rows = GROWS;
    for (int i = c; i < rows; i += 128) sg[i] = batch[base + i];
    __syncthreads();

    const float msc = ms[c];
    float acc = 0.0f;
    float sub = 0.0f;
    int curg = -1;
    for (int r = 0; r < rows; ++r) {
        int g = sg[r];
        if (g != curg) {
            if (curg >= 0) unsafeAtomicAdd(&var_sum[(size_t)curg * DCH + c], acc);
            acc = 0.0f;
            curg = g;
            float cn = cnt[g];
            cn = cn > 1.0f ? cn : 1.0f;
            sub = (mean_sum[(size_t)g * DCH + c] / cn) * msc;
        }
        float o = out[(size_t)(base + r) * DCH + c] - sub;
        acc += o * o;
    }
    if (curg >= 0) unsafeAtomicAdd(&var_sum[(size_t)curg * DCH + c], acc);
}

// ---------------------------------------------------------------------------
// K5: out = gw * (h - mean*ms) / sqrt(var + eps) + gb   (float4 / b128 path)
// Thread owns 4 channels of one node.
// ---------------------------------------------------------------------------
__global__ __launch_bounds__(256) void k_finalize(
    float* __restrict__ out, const int* __restrict__ batch,
    const float* __restrict__ mean_sum, const float* __restrict__ var_sum,
    const float* __restrict__ cnt, const float* __restrict__ ms,
    const float* __restrict__ gw, const float* __restrict__ gb, int N) {
    size_t idx = (size_t)blockIdx.x * 256 + threadIdx.x;   // over N*32 float4s
    size_t total = (size_t)N * (DCH / 4);
    if (idx >= total) return;
    int nrow = (int)(idx >> 5);
    int q    = (int)(idx & 31);          // float4 slot within the row
    int g = batch[nrow];
    float cn = cnt[g];
    cn = cn > 1.0f ? cn : 1.0f;
    float inv_cn = 1.0f / cn;

    float4 m4 = reinterpret_cast<const float4*>(mean_sum + (size_t)g * DCH)[q];
    float4 v4 = reinterpret_cast<const float4*>(var_sum  + (size_t)g * DCH)[q];
    float4 s4 = reinterpret_cast<const float4*>(ms)[q];
    float4 w4 = reinterpret_cast<const float4*>(gw)[q];
    float4 b4 = reinterpret_cast<const float4*>(gb)[q];
    float4 o  = reinterpret_cast<float4*>(out)[idx];

    o.x -= (m4.x * inv_cn) * s4.x;
    o.y -= (m4.y * inv_cn) * s4.y;
    o.z -= (m4.z * inv_cn) * s4.z;
    o.w -= (m4.w * inv_cn) * s4.w;
    o.x = w4.x * o.x / sqrtf(v4.x * inv_cn + GN_EPS) + b4.x;
    o.y = w4.y * o.y / sqrtf(v4.y * inv_cn + GN_EPS) + b4.y;
    o.z = w4.z * o.z / sqrtf(v4.z * inv_cn + GN_EPS) + b4.z;
    o.w = w4.w * o.w / sqrtf(v4.w * inv_cn + GN_EPS) + b4.w;

    reinterpret_cast<float4*>(out)[idx] = o;
}

// ---------------------------------------------------------------------------
extern "C" void kernel_launch(void* const* d_in, const int* in_sizes, int n_in,
                              void* d_out, int out_size, void* d_ws, size_t ws_size,
                              hipStream_t stream) {
    const float* x     = (const float*)d_in[0];
    const int*   eidx  = (const int*)  d_in[1];
    const float* eattr = (const float*)d_in[2];
    const int*   batch = (const int*)  d_in[3];
    const float* W     = (const float*)d_in[4];
    const float* bias  = (const float*)d_in[5];
    const float* gw    = (const float*)d_in[6];
    const float* gb    = (const float*)d_in[7];
    const float* ms    = (const float*)d_in[8];
    float* out = (float*)d_out;

    const int N = in_sizes[0] / DCH;
    const int E = in_sizes[2];

    // Workspace layout
    float* h        = (float*)d_ws;                        // N*DCH
    float* mean_sum = h + (size_t)N * DCH;                 // NGRAPH*DCH
    float* var_sum  = mean_sum + (size_t)NGRAPH * DCH;     // NGRAPH*DCH
    float* cnt      = var_sum + (size_t)NGRAPH * DCH;      // NGRAPH
    float* Wt       = cnt + NGRAPH;                        // DCH*DCH

    // K0: zero agg (d_out) and stats
    {
        size_t n0 = (size_t)N * DCH;
        k_zero_f32<<<(unsigned)((n0 + 255) / 256), 256, 0, stream>>>(out, n0);
        size_t n1 = (size_t)2 * NGRAPH * DCH + NGRAPH;
        k_zero_f32<<<(unsigned)((n1 + 255) / 256), 256, 0, stream>>>(mean_sum, n1);
    }

    // K-pre: transpose W (64KB, one-time)
    k_transpose_w<<<64, 256, 0, stream>>>(W, Wt);

    // K1: WMMA GEMM
    k_gemm_wmma<<<(N + TILE_M - 1) / TILE_M, 256, 0, stream>>>(x, Wt, h, N);

    // K2: edge gather/scale/scatter (src = eidx[0:E], dst = eidx[E:2E])
    {
        int edges_per_block = 8 * EDGES_PER_WAVE;
        k_edge_scatter<<<(E + edges_per_block - 1) / edges_per_block, 256, 0, stream>>>(
            eidx, eidx + E, eattr, h, out, E);
    }

    // K3: bias + ReLU + per-graph mean sums & counts
    k_bias_relu_mean<<<(N + GROWS - 1) / GROWS, 128, 0, stream>>>(
        out, bias, batch, mean_sum, cnt, N);

    // K4: per-graph variance sums
    k_var_pass<<<(N + GROWS - 1) / GROWS, 128, 0, stream>>>(
        out, batch, mean_sum, cnt, ms, var_sum, N);

    // K5: normalize (float4)
    {
        size_t total = (size_t)N * (DCH / 4);
        k_finalize<<<(unsigned)((total + 255) / 256), 256, 0, stream>>>(
            out, batch, mean_sum, var_sum, cnt, ms, gw, gb, N);
    }
}